// SnakeFastLoss_53618371723452
// MI455X (gfx1250) — compile-verified
//
#include <hip/hip_runtime.h>
#include <math.h>

// ---------------------------------------------------------------------------
// Problem constants (from reference)
// ---------------------------------------------------------------------------
#define BB      4
#define HH      384
#define WW      384
#define NN      64
#define HWSZ    (HH*WW)
#define RR      6           // ceil(3*stdev), stdev=2
#define KK      13          // filter size 2*RR+1
#define STEPSZ_ 0.1f
#define ALPHA_  0.01f
#define BETA_   0.01f
#define NSTEPS_ 20
#define NSTEPSW_ 10
#define DMAX_   15.0f
#define EXTF_   10.0f
#define NEGW_   0.5f
#define ENH_    4.0f

#define SEG_STRIDE 16       // padded per-segment record (10 used)
#define CHUNKS     288      // blockIdx.y extent of loss kernel
#define GPW        4        // 16-pixel groups per wave: 288*8*4 = 9216 = HW/16
#define NPART      (BB*CHUNKS)

typedef __attribute__((ext_vector_type(2))) float v2f;
typedef __attribute__((ext_vector_type(8))) float v8f;
typedef __attribute__((__vector_size__(4 * sizeof(int)))) int v4i;

// Feature probes ------------------------------------------------------------
#if defined(__has_builtin)
#if __has_builtin(__builtin_amdgcn_wmma_f32_16x16x4_f32)
#define HAVE_WMMA_F32X4 1
#endif
#if __has_builtin(__builtin_amdgcn_global_load_async_to_lds_b128) && \
    __has_builtin(__builtin_amdgcn_s_wait_asynccnt)
#define HAVE_ASYNC_LDS 1
#endif
#if __has_builtin(__builtin_amdgcn_sqrtf)
#define HAVE_FAST_SQRT 1
#endif
#endif

#ifdef HAVE_ASYNC_LDS
typedef v4i __attribute__((address_space(1)))* async_gptr_t;  // global src
typedef v4i __attribute__((address_space(3)))* async_lptr_t;  // LDS dst
#endif

// Raw v_sqrt_f32 (~1 ulp) — value feeds only a min/compare, precision is ample.
__device__ __forceinline__ float fast_sqrtf(float x) {
#ifdef HAVE_FAST_SQRT
    return __builtin_amdgcn_sqrtf(x);
#else
    return sqrtf(x);
#endif
}

// ---------------------------------------------------------------------------
// Kernel 1: 13x13 Gaussian-derivative conv (cross-correlation, SAME/zero pad)
// produces gimg = conv(pred)*10 and gimgW = conv(|pred|)*10, 2 channels each.
// 16x16 output tile per block, input tile (+halo) and filter staged in LDS.
// ---------------------------------------------------------------------------
__global__ __launch_bounds__(256) void conv_kernel(const float* __restrict__ pred,
                                                   const float* __restrict__ fltr,
                                                   float* __restrict__ gimg,
                                                   float* __restrict__ gimgW) {
    __shared__ float sv[28*28];
    __shared__ float sa[28*28];
    __shared__ float sf[2*KK*KK];
    int b   = blockIdx.z;
    int ty0 = blockIdx.y*16 - RR;
    int tx0 = blockIdx.x*16 - RR;
    int tid = threadIdx.x;
    const float* p = pred + (size_t)b*HWSZ;
    for (int idx = tid; idx < 28*28; idx += 256) {
        int ly = idx / 28, lx = idx % 28;
        int y = ty0 + ly, x = tx0 + lx;
        float v = 0.f;
        if (y >= 0 && y < HH && x >= 0 && x < WW) v = p[y*WW + x];
        sv[idx] = v;
        sa[idx] = fabsf(v);
    }
    for (int idx = tid; idx < 2*KK*KK; idx += 256) sf[idx] = fltr[idx];
    __syncthreads();

    int oy = tid >> 4, ox = tid & 15;
    float a0 = 0.f, a1 = 0.f, a2 = 0.f, a3 = 0.f;
    for (int i = 0; i < KK; i++) {
        const float* rv = &sv[(oy+i)*28 + ox];
        const float* ra = &sa[(oy+i)*28 + ox];
        const float* f0 = &sf[i*KK];
        const float* f1 = &sf[KK*KK + i*KK];
        #pragma unroll
        for (int j = 0; j < KK; j++) {
            float v = rv[j], av = ra[j], c0 = f0[j], c1 = f1[j];
            a0 = fmaf(v,  c0, a0);
            a1 = fmaf(v,  c1, a1);
            a2 = fmaf(av, c0, a2);
            a3 = fmaf(av, c1, a3);
        }
    }
    int y = blockIdx.y*16 + oy, x = blockIdx.x*16 + ox;
    size_t o = (size_t)b*2*HWSZ + (size_t)y*WW + x;
    gimg [o]        = a0*EXTF_;
    gimg [o + HWSZ] = a1*EXTF_;
    gimgW[o]        = a2*EXTF_;
    gimgW[o + HWSZ] = a3*EXTF_;
}

// ---------------------------------------------------------------------------
// Kernel 2: snake iteration (one block per batch, thread n = node n),
// then emit per-segment records for the distance-map kernel.
// ---------------------------------------------------------------------------
__device__ __forceinline__ float bilerp1(const float* img, float y, float x) {
    y = fminf(fmaxf(y, 0.f), (float)(HH-1));
    x = fminf(fmaxf(x, 0.f), (float)(WW-1));
    int y0 = (int)floorf(y); y0 = min(max(y0, 0), HH-2);
    int x0 = (int)floorf(x); x0 = min(max(x0, 0), WW-2);
    float wy = y - (float)y0, wx = x - (float)x0;
    const float* r = img + y0*WW + x0;
    float v00 = r[0], v01 = r[1], v10 = r[WW], v11 = r[WW+1];
    return v00*(1.f-wy)*(1.f-wx) + v01*(1.f-wy)*wx
         + v10*wy*(1.f-wx)       + v11*wy*wx;
}

__global__ __launch_bounds__(64) void snake_kernel(const float* __restrict__ nodes,
                                                   const float* __restrict__ widths,
                                                   const float* __restrict__ gimg,
                                                   const float* __restrict__ gimgW,
                                                   float* __restrict__ seg) {
    __shared__ float py[NN], px[NN], qy[NN], qx[NN], wv[NN];
    int b = blockIdx.x;
    int n = threadIdx.x;
    const float* g0  = gimg  + (size_t)b*2*HWSZ;
    const float* g1  = g0 + HWSZ;
    const float* gw0 = gimgW + (size_t)b*2*HWSZ;
    const float* gw1 = gw0 + HWSZ;

    py[n] = nodes[((size_t)b*NN + n)*2 + 0];
    px[n] = nodes[((size_t)b*NN + n)*2 + 1];
    wv[n] = widths[(size_t)b*NN + n];
    __syncthreads();

    int nm = max(n-1, 0), np = min(n+1, NN-1);
    for (int s = 0; s < NSTEPS_; s++) {
        // d2 with replicated ends
        float d2y = py[nm] - 2.f*py[n] + py[np];
        float d2x = px[nm] - 2.f*px[n] + px[np];
        qy[n] = d2y; qx[n] = d2x;
        __syncthreads();
        float d4y = qy[nm] - 2.f*qy[n] + qy[np];
        float d4x = qx[nm] - 2.f*qx[n] + qx[np];
        float fy = bilerp1(g0, py[n], px[n]);
        float fx = bilerp1(g1, py[n], px[n]);
        float ny2 = py[n] + STEPSZ_*(fy + ALPHA_*d2y - BETA_*d4y);
        float nx2 = px[n] + STEPSZ_*(fx + ALPHA_*d2x - BETA_*d4x);
        __syncthreads();
        py[n] = fminf(fmaxf(ny2, 0.f), (float)(HH-1));
        px[n] = fminf(fmaxf(nx2, 0.f), (float)(WW-1));
        __syncthreads();
    }

    float w  = wv[n];
    float yy = py[n], xx = px[n];
    for (int s = 0; s < NSTEPSW_; s++) {
        float gw = bilerp1(gw0, yy, xx) + bilerp1(gw1, yy, xx);
        w = fminf(fmaxf(w + STEPSZ_*gw, 0.5f), DMAX_);
    }
    wv[n] = w;
    __syncthreads();

    // segment record: a_y,a_x,ab_y,ab_x,ab2,inv_ab2,c1=a.ab,na2=|a|^2,wa,dw
    float* sp = seg + ((size_t)b*NN + n)*SEG_STRIDE;
    if (n < NN-1) {
        float ay = py[n],   ax = px[n];
        float by = py[n+1], bx = px[n+1];
        float aby = by - ay, abx = bx - ax;
        float ab2 = aby*aby + abx*abx + 1e-8f;
        sp[0] = ay;  sp[1] = ax;  sp[2] = aby; sp[3] = abx;
        sp[4] = ab2; sp[5] = 1.f/ab2;
        sp[6] = ay*aby + ax*abx;
        sp[7] = ay*ay + ax*ax;
        sp[8] = wv[n]; sp[9] = wv[n+1] - wv[n];
    } else {
        // sentinel (padding 63->64): pushes distance to ~1e8, never wins min
        sp[0] = 1e8f; sp[1] = 1e8f; sp[2] = 0.f; sp[3] = 0.f;
        sp[4] = 1.f;  sp[5] = 0.f;  sp[6] = 0.f; sp[7] = 2e16f;
        sp[8] = 0.f;  sp[9] = 0.f;
    }
    // pad lanes [10..15]: keep deterministic content
    sp[10] = 0.f; sp[11] = 0.f; sp[12] = 0.f; sp[13] = 0.f; sp[14] = 0.f; sp[15] = 0.f;
}

// ---------------------------------------------------------------------------
// Kernel 3: distance map + weighted MSE via WMMA.
// Per wave: A = 16 pixels x K4 ([y,x,0,0]); for each 16-segment tile,
// D0 = pts.ab, D1 = pts.a via V_WMMA_F32_16X16X4_F32; epilogue computes
// point-segment distance minus interpolated width; min over segments via
// per-lane running min + shfl_xor butterfly; then per-pixel loss.
// Segment table (4KB/block) staged to LDS via async global->LDS DMA when
// the toolchain exposes the gfx1250 async builtins.
// ---------------------------------------------------------------------------
__global__ __launch_bounds__(256) void dmap_loss_kernel(const float* __restrict__ pred,
                                                        const float* __restrict__ seg,
                                                        float* __restrict__ partial) {
    __shared__ float ssg[NN*SEG_STRIDE];   // 64 segs x 16 floats = 4KB
    __shared__ float red[256];
    int b   = blockIdx.x;
    int tid = threadIdx.x;

#ifdef HAVE_ASYNC_LDS
    {
        // 256 threads x 16B = 4KB, exactly the segment table.
        const float* gsrc = seg + (size_t)b*NN*SEG_STRIDE + (size_t)tid*4;
        __builtin_amdgcn_global_load_async_to_lds_b128(
            (async_gptr_t)(void*)const_cast<float*>(gsrc),
            (async_lptr_t)(void*)&ssg[tid*4],
            0, 0);
        __builtin_amdgcn_s_wait_asynccnt(0);
    }
    __syncthreads();
#else
    for (int idx = tid; idx < NN*SEG_STRIDE; idx += 256)
        ssg[idx] = seg[(size_t)b*NN*SEG_STRIDE + idx];
    __syncthreads();
#endif

    int lane = tid & 31;
    int wvid = tid >> 5;
    int half = lane >> 4;    // C/D layout: lanes 16-31 hold M=8..15
    int nlo  = lane & 15;    // N within tile
    const float* pb = pred + (size_t)b*HWSZ;
    float acc = 0.f;

    int waveid = blockIdx.y*8 + wvid;
    for (int i = 0; i < GPW; i++) {
        int g   = waveid*GPW + i;         // 16-pixel group; W%16==0 => one row
        int gy  = g / (WW/16);
        int gx0 = (g % (WW/16))*16;
        float fy = (float)gy;

        // A (16x4 f32): lanes 0-15 carry K=0,1 = (y, x_M); lanes 16-31 carry K=2,3 = 0
        v2f A;
        A.x = half ? 0.f : fy;
        A.y = half ? 0.f : (float)(gx0 + lane);

        float minv[8];
        #pragma unroll
        for (int v = 0; v < 8; v++) minv[v] = 1e30f;

        for (int t = 0; t < 4; t++) {
            const float* sg = &ssg[(t*16 + nlo)*SEG_STRIDE];
            float ay  = sg[0], ax  = sg[1], aby = sg[2], abx = sg[3];
            float ab2 = sg[4], inv = sg[5], c1  = sg[6], na2 = sg[7];
            float wa  = sg[8], dw  = sg[9];

            // B (4x16 f32): lanes 0-15 rows K=0,1; lanes 16-31 rows K=2,3 (zero)
            v2f Bb, Ba;
            Bb.x = half ? 0.f : aby;  Bb.y = half ? 0.f : abx;
            Ba.x = half ? 0.f : ay;   Ba.y = half ? 0.f : ax;

            v8f cz = {0.f,0.f,0.f,0.f,0.f,0.f,0.f,0.f};
            v8f dotb, dota;
#ifdef HAVE_WMMA_F32X4
            dotb = __builtin_amdgcn_wmma_f32_16x16x4_f32(
                false, A, false, Bb, (short)0, cz, false, false);
            dota = __builtin_amdgcn_wmma_f32_16x16x4_f32(
                false, A, false, Ba, (short)0, cz, false, false);
#else
            #pragma unroll
            for (int v = 0; v < 8; v++) {
                float xm = (float)(gx0 + v + 8*half);
                dotb[v] = fy*aby + xm*abx;
                dota[v] = fy*ay  + xm*ax;
            }
#endif
            // epilogue: element (M = v + 8*half, N = nlo)
            #pragma unroll
            for (int v = 0; v < 8; v++) {
                float dotv = dotb[v] - c1;                      // pa.ab
                float tt   = fminf(fmaxf(dotv*inv, 0.f), 1.f);  // clamp(dot/ab2)
                float xm   = (float)(gx0 + v + 8*half);
                float pts2 = fy*fy + xm*xm;
                float pa2  = pts2 - 2.f*dota[v] + na2;          // |pts - a|^2
                float dd2  = pa2 - tt*(2.f*dotv - tt*ab2);      // |pa - t*ab|^2
                float dd   = fast_sqrtf(fmaxf(dd2, 0.f) + 1e-12f);
                float val  = dd - (wa + tt*dw);
                minv[v] = fminf(minv[v], val);
            }
        }

        // min over N (16 lanes per half) + per-pixel loss
        #pragma unroll
        for (int v = 0; v < 8; v++) {
            float m = minv[v];
            m = fminf(m, __shfl_xor(m, 1, 32));
            m = fminf(m, __shfl_xor(m, 2, 32));
            m = fminf(m, __shfl_xor(m, 4, 32));
            m = fminf(m, __shfl_xor(m, 8, 32));
            if (nlo == v) {                       // one lane per pixel row M
                int mrow  = v + 8*half;
                float sn  = fminf(m, DMAX_);
                bool  neg = (sn < 0.f);
                float snk = neg ? sn*ENH_ : sn;
                float pv  = pb[gy*WW + gx0 + mrow];
                float df  = pv - snk;
                acc += df*df * (neg ? (1.f + NEGW_) : 1.f);
            }
        }
    }

    red[tid] = acc;
    __syncthreads();
    for (int s = 128; s > 0; s >>= 1) {
        if (tid < s) red[tid] += red[tid + s];
        __syncthreads();
    }
    if (tid == 0) partial[blockIdx.x*gridDim.y + blockIdx.y] = red[0];
}

// ---------------------------------------------------------------------------
// Kernel 4: deterministic final reduction -> mean
// ---------------------------------------------------------------------------
__global__ __launch_bounds__(256) void reduce_kernel(const float* __restrict__ partial,
                                                     float* __restrict__ out) {
    __shared__ float red[256];
    float s = 0.f;
    for (int i = threadIdx.x; i < NPART; i += 256) s += partial[i];
    red[threadIdx.x] = s;
    __syncthreads();
    for (int k = 128; k > 0; k >>= 1) {
        if (threadIdx.x < k) red[threadIdx.x] += red[threadIdx.x + k];
        __syncthreads();
    }
    if (threadIdx.x == 0) out[0] = red[0] / (float)((size_t)BB*HWSZ);
}

// ---------------------------------------------------------------------------
extern "C" void kernel_launch(void* const* d_in, const int* in_sizes, int n_in,
                              void* d_out, int out_size, void* d_ws, size_t ws_size,
                              hipStream_t stream) {
    const float* pred   = (const float*)d_in[0];   // [B,1,H,W]
    const float* nodes  = (const float*)d_in[1];   // [B,N,2]
    const float* widths = (const float*)d_in[2];   // [B,N]
    const float* fltr   = (const float*)d_in[3];   // [2,1,13,13]

    float* ws    = (float*)d_ws;
    float* gimg  = ws;                               // B*2*HW
    float* gimgW = gimg  + (size_t)BB*2*HWSZ;        // B*2*HW
    float* seg   = gimgW + (size_t)BB*2*HWSZ;        // B*64*16
    float* part  = seg   + (size_t)BB*NN*SEG_STRIDE; // NPART

    conv_kernel<<<dim3(WW/16, HH/16, BB), 256, 0, stream>>>(pred, fltr, gimg, gimgW);
    snake_kernel<<<dim3(BB), 64, 0, stream>>>(nodes, widths, gimg, gimgW, seg);
    dmap_loss_kernel<<<dim3(BB, CHUNKS), 256, 0, stream>>>(pred, seg, part);
    reduce_kernel<<<1, 256, 0, stream>>>(part, (float*)d_out);
}